// GraphSAGE_9663676416699
// MI455X (gfx1250) — compile-verified
//
#include <hip/hip_runtime.h>
#include <hip/hip_bf16.h>

// GraphSAGE inference for MI455X (gfx1250, wave32).
//   h1 = relu(mean1 @ Wl1 + x @ Wr1 + b1)
//   h2 = mean2 @ Wl2 + h1 @ Wr2 + b2          -> d_out[0 : N*128)
//   logits = relu(h2 @ Wc1 + bc1) @ Wc2 + bc2 -> d_out[N*128 : N*130)
// mean{1,2} built by per-edge wave32 scatter with global f32 atomics (L2-resident),
// dense layers via V_WMMA_F32_16X16X4_F32 (full fp32; workload is memory-bound, so
// low-precision WMMA buys nothing and f32 matches reference numerics).
// GEMM uses double-buffered LDS weight panels + whole-panel A prefetch into VGPRs
// so global-load latency overlaps the 64-WMMA compute phase of each panel.

typedef __attribute__((ext_vector_type(2))) float v2f;
typedef __attribute__((ext_vector_type(8))) float v8f;

#define TILE_M 128          // rows per block (8 waves x 16)
#define TILE_N 64           // cols per block (4 wmma tiles of 16)
#define BSTR   68           // LDS row stride in floats (64 + 4 pad -> no bank conflicts)

// ---------------------------------------------------------------- utility ---
__global__ void zero_kernel(float* __restrict__ p, long long n) {
    long long i = (long long)blockIdx.x * blockDim.x + threadIdx.x;
    long long stride = (long long)gridDim.x * blockDim.x;
    long long n4 = n >> 2;
    const float4 z = make_float4(0.f, 0.f, 0.f, 0.f);
    for (long long j = i; j < n4; j += stride) ((float4*)p)[j] = z;
    for (long long j = (n4 << 2) + i; j < n; j += stride) p[j] = 0.0f;
}

__global__ void degree_kernel(const int* __restrict__ dst, float* __restrict__ cnt, int nedges) {
    int e = blockIdx.x * blockDim.x + threadIdx.x;
    if (e < nedges) atomicAdd(cnt + dst[e], 1.0f);
}

// one wave32 per edge; lane covers 4 floats per 128-feature chunk
__global__ __launch_bounds__(256)
void scatter_add_kernel(const float* __restrict__ feat, const int* __restrict__ src,
                        const int* __restrict__ dst, float* __restrict__ agg,
                        int nedges, int D) {
    int gw   = (blockIdx.x * 256 + threadIdx.x) >> 5;   // global wave id == edge id
    int lane = threadIdx.x & 31;
    if (gw >= nedges) return;
    int s = src[gw];               // uniform within wave -> scalarized
    int d = dst[gw];
    const float* fs = feat + (size_t)s * D;
    float*       ad = agg  + (size_t)d * D;
    for (int base = lane * 4; base < D; base += 128) {
        float4 v = *(const float4*)(fs + base);
        atomicAdd(ad + base + 0, v.x);
        atomicAdd(ad + base + 1, v.y);
        atomicAdd(ad + base + 2, v.z);
        atomicAdd(ad + base + 3, v.w);
    }
}

// ------------------------------------------------------------- WMMA GEMM ---
// out[nrows,Nc] = act( scale(A1) @ W1  (+ A2 @ W2)  + bias )
//   scale(A1) = A1 / max(cnt,1) per row when cnt != null (SAGE mean aggregation)
//   A2 == null -> single-GEMM mode (classifier hidden layer)
// Block: 256 threads = 8 waves; each wave owns a 16(M) x 64(N) strip.
// K is processed in 64-wide panels; panels of pass 0 (A1,W1) then pass 1 (A2,W2).
__global__ __launch_bounds__(256)
void sage_gemm_wmma(const float* __restrict__ A1, const float* __restrict__ W1,
                    const float* __restrict__ cnt,
                    const float* __restrict__ A2, const float* __restrict__ W2,
                    const float* __restrict__ bias,
                    float* __restrict__ out,
                    int nrows, int K, int Nc, int do_relu) {
    __shared__ float smemB[2][64 * BSTR];         // double-buffered 64x64 weight panel

    const int tid  = threadIdx.x;
    const int lane = tid & 31;
    const int wave = tid >> 5;
    const int half = lane >> 4;                   // 0 or 1
    const int mlan = lane & 15;

    const int colBase = blockIdx.x * TILE_N;
    const int rowBase = blockIdx.y * TILE_M;

    // A-fragment row for this lane is constant across the whole K loop
    const int rowA  = rowBase + wave * 16 + mlan;
    const int rowAc = rowA < nrows ? rowA : (nrows - 1);   // clamp (keep EXEC full)

    float scale1 = 1.0f;
    if (cnt) scale1 = 1.0f / fmaxf(cnt[rowAc], 1.0f);

    // cooperative staging geometry (fixed per thread): thread covers float4 column
    // c4 of k-rows rbase, rbase+16, rbase+32, rbase+48 of the 64x64 panel
    const int c4    = tid & 15;
    const int rbase = tid >> 4;

    const int panelsPerPass = K >> 6;
    const int nPanels = panelsPerPass * ((A2 != nullptr) ? 2 : 1);

    auto wLoad = [&](int p, float4* w) {          // issue 4 global loads (batched)
        int pass = (p >= panelsPerPass) ? 1 : 0;
        int kb   = (p - pass * panelsPerPass) << 6;
        const float* W  = pass ? W2 : W1;
        const float* Wp = W + (size_t)kb * Nc + colBase;
        #pragma unroll
        for (int i = 0; i < 4; ++i)
            w[i] = *(const float4*)(Wp + (size_t)(rbase + 16 * i) * Nc + c4 * 4);
    };
    auto wStore = [&](int buf, const float4* w) { // 4 ds_store_b128
        #pragma unroll
        for (int i = 0; i < 4; ++i)
            *(float4*)(&smemB[buf][(rbase + 16 * i) * BSTR + c4 * 4]) = w[i];
    };
    auto aLoad = [&](int p, float2* af) {         // whole 64-wide A panel, 16 b64 loads
        int pass = (p >= panelsPerPass) ? 1 : 0;
        int kb   = (p - pass * panelsPerPass) << 6;
        const float* A  = pass ? A2 : A1;
        const float* Ap = A + (size_t)rowAc * K + kb + 2 * half;
        #pragma unroll
        for (int i = 0; i < 16; ++i)
            af[i] = *(const float2*)(Ap + i * 4);
    };

    v8f acc[4];
    #pragma unroll
    for (int t = 0; t < 4; ++t) acc[t] = (v8f){0,0,0,0,0,0,0,0};

    { // prologue: stage panel 0 into buffer 0
        float4 w0[4];
        wLoad(0, w0);
        wStore(0, w0);
    }
    __syncthreads();

    for (int p = 0; p < nPanels; ++p) {
        const int  buf  = p & 1;
        const bool more = (p + 1) < nPanels;

        float2 af[16];
        aLoad(p, af);                 // A panel in flight first (consumed progressively)
        float4 wn[4];
        if (more) wLoad(p + 1, wn);   // next W panel in flight during compute

        const float s = (p >= panelsPerPass) ? 1.0f : scale1;
        #pragma unroll
        for (int kk = 0; kk < 16; ++kk) {
            const int k0 = kk * 4 + 2 * half;     // per-lane k pair (ISA f32 A layout)
            v2f a; a[0] = af[kk].x * s; a[1] = af[kk].y * s;
            #pragma unroll
            for (int t = 0; t < 4; ++t) {
                v2f b;
                b[0] = smemB[buf][(k0    ) * BSTR + t * 16 + mlan];
                b[1] = smemB[buf][(k0 + 1) * BSTR + t * 16 + mlan];
                acc[t] = __builtin_amdgcn_wmma_f32_16x16x4_f32(
                    /*neg_a=*/false, a, /*neg_b=*/false, b,
                    /*c_mod=*/(short)0, acc[t], /*reuse_a=*/false, /*reuse_b=*/false);
            }
        }
        if (more) wStore(buf ^ 1, wn);            // fill the other buffer
        __syncthreads();                          // one barrier per panel
    }

    // epilogue: bias + optional relu, store with D-matrix layout (m = v + 8*half)
    #pragma unroll
    for (int t = 0; t < 4; ++t) {
        int   n  = colBase + t * 16 + mlan;
        float bv = bias ? bias[n] : 0.0f;
        #pragma unroll
        for (int v = 0; v < 8; ++v) {
            int row = rowBase + wave * 16 + v + 8 * half;
            if (row < nrows) {
                float r = acc[t][v] + bv;
                if (do_relu) r = fmaxf(r, 0.0f);
                out[(size_t)row * Nc + n] = r;
            }
        }
    }
}

// ------------------------------------------------------- 2-wide classifier ---
__global__ void logits_kernel(const float* __restrict__ mlp, const float* __restrict__ Wc2,
                              const float* __restrict__ bc2, float* __restrict__ logits,
                              int nrows, int K) {
    int node = blockIdx.x * blockDim.x + threadIdx.x;
    if (node >= nrows) return;
    const float* m = mlp + (size_t)node * K;
    float a0 = 0.0f, a1 = 0.0f;
    for (int k = 0; k < K; k += 4) {
        float4 v = *(const float4*)(m + k);
        float2 w0 = *(const float2*)(Wc2 + (k + 0) * 2);  // uniform -> scalar loads
        float2 w1 = *(const float2*)(Wc2 + (k + 1) * 2);
        float2 w2 = *(const float2*)(Wc2 + (k + 2) * 2);
        float2 w3 = *(const float2*)(Wc2 + (k + 3) * 2);
        a0 += v.x * w0.x + v.y * w1.x + v.z * w2.x + v.w * w3.x;
        a1 += v.x * w0.y + v.y * w1.y + v.z * w2.y + v.w * w3.y;
    }
    logits[(size_t)node * 2 + 0] = a0 + bc2[0];
    logits[(size_t)node * 2 + 1] = a1 + bc2[1];
}

// ------------------------------------------------------------------ driver ---
extern "C" void kernel_launch(void* const* d_in, const int* in_sizes, int n_in,
                              void* d_out, int out_size, void* d_ws, size_t ws_size,
                              hipStream_t stream) {
    const float* x   = (const float*)d_in[0];
    const int*   ei  = (const int*)d_in[1];
    const float* Wl1 = (const float*)d_in[2];
    const float* Wr1 = (const float*)d_in[3];
    const float* b1  = (const float*)d_in[4];
    const float* Wl2 = (const float*)d_in[5];
    const float* Wr2 = (const float*)d_in[6];
    const float* b2  = (const float*)d_in[7];
    const float* Wc1 = (const float*)d_in[8];
    const float* bc1 = (const float*)d_in[9];
    const float* Wc2 = (const float*)d_in[10];
    const float* bc2 = (const float*)d_in[11];

    const int IN = 128, HID = 256, OUT = 128;
    const int n  = in_sizes[0] / IN;       // 100000
    const int ne = in_sizes[1] / 2;        // 1600000
    const int* src = ei;
    const int* dst = ei + ne;

    // workspace layout (floats): cnt | agg1 | agg2 | h1   (agg2 reused as mlp hidden)
    float* cnt  = (float*)d_ws;
    float* agg1 = cnt  + n;
    float* agg2 = agg1 + (size_t)n * IN;
    float* h1   = agg2 + (size_t)n * HID;
    float* mlp  = agg2;                    // safe: agg2 dead after layer-2 GEMM
    float* h2     = (float*)d_out;                     // [n, OUT]
    float* logits = (float*)d_out + (size_t)n * OUT;   // [n, 2]

    // 1. zero accumulation buffers (cnt + agg1 + agg2) -- required every call
    long long zn = (long long)n * (1 + IN + HID);
    zero_kernel<<<2048, 256, 0, stream>>>(cnt, zn);

    // 2. in-degree counts
    degree_kernel<<<(ne + 255) / 256, 256, 0, stream>>>(dst, cnt, ne);

    // 3. layer-1 neighbor scatter: agg1[dst] += x[src]
    scatter_add_kernel<<<(ne + 7) / 8, 256, 0, stream>>>(x, src, dst, agg1, ne, IN);

    // 4. h1 = relu(mean1 @ Wl1 + x @ Wr1 + b1)
    dim3 g1(HID / TILE_N, (n + TILE_M - 1) / TILE_M);
    sage_gemm_wmma<<<g1, 256, 0, stream>>>(agg1, Wl1, cnt, x, Wr1, b1, h1, n, IN, HID, 1);

    // 5. layer-2 neighbor scatter: agg2[dst] += h1[src]
    scatter_add_kernel<<<(ne + 7) / 8, 256, 0, stream>>>(h1, src, dst, agg2, ne, HID);

    // 6. h2 = mean2 @ Wl2 + h1 @ Wr2 + b2   (node embeddings -> d_out)
    dim3 g2(OUT / TILE_N, (n + TILE_M - 1) / TILE_M);
    sage_gemm_wmma<<<g2, 256, 0, stream>>>(agg2, Wl2, cnt, h1, Wr2, b2, h2, n, HID, OUT, 0);

    // 7. mlp = relu(h2 @ Wc1 + bc1)   (single-GEMM mode)
    dim3 g3(HID / TILE_N, (n + TILE_M - 1) / TILE_M);
    sage_gemm_wmma<<<g3, 256, 0, stream>>>(h2, Wc1, nullptr, nullptr, nullptr, bc1, mlp, n, OUT, HID, 1);

    // 8. logits = mlp @ Wc2 + bc2  (Nc=2: plain dot kernel, WMMA tile would waste 8x)
    logits_kernel<<<(n + 255) / 256, 256, 0, stream>>>(mlp, Wc2, bc2, logits, n, HID);
}